// GCN_5025111736960
// MI455X (gfx1250) — compile-verified
//
#include <hip/hip_runtime.h>

// GCN 2-layer forward for MI455X (gfx1250).
// GEMM uses V_WMMA_F32_16X16X4_F32 (wave32, 16x16 tile per wave, K=64 -> 16 WMMAs).
// Aggregation uses per-edge wave parallelism with global_atomic_add_f32 into an
// accumulator that fits in the 192MB L2.

typedef float v2f __attribute__((ext_vector_type(2)));
typedef float v8f __attribute__((ext_vector_type(8)));

#define FD 64  // feature dim

// ---- degree / normalization ------------------------------------------------

__global__ void k_fill_deg(float* __restrict__ deg, int n) {
    int i = blockIdx.x * blockDim.x + threadIdx.x;
    if (i < n) deg[i] = 1.0f;  // self-loop weight
}

__global__ void k_deg_accum(const long long* __restrict__ ei,
                            const float* __restrict__ ew,
                            float* __restrict__ deg, int E) {
    int e = blockIdx.x * blockDim.x + threadIdx.x;
    if (e < E) {
        long long d = ei[(size_t)E + e];  // dst row
        atomicAdd(&deg[d], ew[e]);
    }
}

__global__ void k_dis(float* __restrict__ deg, int n) {
    int i = blockIdx.x * blockDim.x + threadIdx.x;
    if (i < n) {
        float d = deg[i];
        deg[i] = (d > 0.0f) ? rsqrtf(d) : 0.0f;  // in-place -> dis
    }
}

// ---- WMMA fp32 GEMM: H[n,64] = X[n,64] @ W[64,64] --------------------------
// One wave per 16x16 output tile. A fragment layout (ISA 7.12.2, 32-bit A 16x4):
//   lane L (m = L&15, kh = (L>>4)*2): a[0]=A[m][k+kh], a[1]=A[m][k+kh+1]
// B/D row-striped across lanes: D vgpr r -> row (lane<16 ? r : r+8), col lane&15.

__global__ void k_gemm_wmma(const float* __restrict__ X,
                            const float* __restrict__ W,
                            float* __restrict__ H, int n) {
    int wave = (blockIdx.x * blockDim.x + threadIdx.x) >> 5;
    int lane = threadIdx.x & 31;
    const int tilesN = FD / 16;          // 4 column tiles
    int tileM = wave / tilesN;
    int tileN = wave - tileM * tilesN;
    int row0 = tileM * 16;
    if (row0 >= n) return;               // wave-uniform: EXEC stays all-1s

    int m  = lane & 15;
    int kh = (lane >> 4) * 2;            // 0 or 2
    int col = tileN * 16 + m;

    int arow = row0 + m;
    if (arow >= n) arow = n - 1;         // clamp (n is a multiple of 16 here)
    const float* ap = X + (size_t)arow * FD;

    v8f acc = {0.f, 0.f, 0.f, 0.f, 0.f, 0.f, 0.f, 0.f};
    #pragma unroll
    for (int k = 0; k < FD; k += 4) {
        v2f a, b;
        a[0] = ap[k + kh];
        a[1] = ap[k + kh + 1];
        b[0] = W[(size_t)(k + kh)     * FD + col];
        b[1] = W[(size_t)(k + kh + 1) * FD + col];
        acc = __builtin_amdgcn_wmma_f32_16x16x4_f32(
            /*neg_a=*/false, a, /*neg_b=*/false, b,
            /*c_mod=*/(short)0, acc, /*reuse_a=*/false, /*reuse_b=*/false);
    }

    int mBase = row0 + (lane >> 4) * 8;  // lanes 16-31 hold rows +8
    #pragma unroll
    for (int r = 0; r < 8; ++r) {
        int row = mBase + r;
        if (row < n) H[(size_t)row * FD + col] = acc[r];
    }
}

// ---- aggregation -----------------------------------------------------------

// agg[i,:] = dis[i]^2 * h[i,:]   (self-loop term, also zero-initializes agg)
__global__ void k_self_init(const float* __restrict__ dis,
                            const float* __restrict__ h,
                            float* __restrict__ agg, int total) {
    int t = blockIdx.x * blockDim.x + threadIdx.x;
    if (t < total) {
        float s = dis[t >> 6];
        agg[t] = s * s * h[t];
    }
}

// One wave per edge; lane j handles features 2j, 2j+1.
// agg[dst,:] += (dis[src]*ew*dis[dst]) * h[src,:]
__global__ void k_edge_scatter(const long long* __restrict__ ei,
                               const float* __restrict__ ew,
                               const float* __restrict__ dis,
                               const float* __restrict__ h,
                               float* __restrict__ agg, int E) {
    int t = blockIdx.x * blockDim.x + threadIdx.x;
    int e = t >> 5;
    int lane = t & 31;
    if (e >= E) return;
    long long s = ei[e];
    long long d = ei[(size_t)E + e];
    float c = dis[s] * ew[e] * dis[d];
    const float2* hv = (const float2*)(h + (size_t)s * FD);
    float2 v = hv[lane];
    float* out = agg + (size_t)d * FD + lane * 2;
    atomicAdd(out,     c * v.x);
    atomicAdd(out + 1, c * v.y);
}

// ---- epilogues -------------------------------------------------------------

__global__ void k_bias_relu(const float* __restrict__ agg,
                            const float* __restrict__ b,
                            float* __restrict__ out, int total) {
    int t = blockIdx.x * blockDim.x + threadIdx.x;
    if (t < total) {
        float v = agg[t] + b[t & 63];
        out[t] = v > 0.0f ? v : 0.0f;
    }
}

// out = (x1 + relu(agg + b2)) * 0.5, with x1 already resident in `out`.
__global__ void k_final(const float* __restrict__ agg,
                        const float* __restrict__ b,
                        float* __restrict__ out, int total) {
    int t = blockIdx.x * blockDim.x + threadIdx.x;
    if (t < total) {
        float v = agg[t] + b[t & 63];
        v = v > 0.0f ? v : 0.0f;
        out[t] = (out[t] + v) * 0.5f;
    }
}

// ---- launch ----------------------------------------------------------------

static inline int cdiv_ll(long long a, long long b) { return (int)((a + b - 1) / b); }

extern "C" void kernel_launch(void* const* d_in, const int* in_sizes, int n_in,
                              void* d_out, int out_size, void* d_ws, size_t ws_size,
                              hipStream_t stream) {
    (void)n_in; (void)out_size; (void)ws_size;
    const float*     x  = (const float*)d_in[0];
    const long long* ei = (const long long*)d_in[1];   // [2,E] int64: src row then dst row
    const float*     ew = (const float*)d_in[2];
    const float*     W1 = (const float*)d_in[3];
    const float*     b1 = (const float*)d_in[4];
    const float*     W2 = (const float*)d_in[5];
    const float*     b2 = (const float*)d_in[6];
    float*           out = (float*)d_out;

    int n = in_sizes[0] / FD;
    int E = in_sizes[2];
    int total = n * FD;

    // workspace: dis[n] | h[n*64] | agg[n*64]  (~52 MB)
    float* dis = (float*)d_ws;
    size_t nAlign = ((size_t)n + 63) & ~(size_t)63;
    float* h   = dis + nAlign;
    float* agg = h + (size_t)n * FD;

    const int B = 256;
    int gN  = cdiv_ll(n, B);
    int gE  = cdiv_ll(E, B);
    int gT  = cdiv_ll(total, B);
    int gSc = cdiv_ll((long long)E * 32, B);
    long long waves = (long long)cdiv_ll(n, 16) * (FD / 16);
    int gGemm = cdiv_ll(waves * 32, B);

    // normalization: deg -> dis (in place)
    k_fill_deg <<<gN, B, 0, stream>>>(dis, n);
    k_deg_accum<<<gE, B, 0, stream>>>(ei, ew, dis, E);
    k_dis      <<<gN, B, 0, stream>>>(dis, n);

    // layer 1: x1 = relu(aggregate(x @ W1) + b1)  -> stored in d_out
    k_gemm_wmma   <<<gGemm, B, 0, stream>>>(x, W1, h, n);
    k_self_init   <<<gT,    B, 0, stream>>>(dis, h, agg, total);
    k_edge_scatter<<<gSc,   B, 0, stream>>>(ei, ew, dis, h, agg, E);
    k_bias_relu   <<<gT,    B, 0, stream>>>(agg, b1, out, total);

    // layer 2: x2 = relu(aggregate(x1 @ W2) + b2); out = (x1 + x2) * 0.5
    k_gemm_wmma   <<<gGemm, B, 0, stream>>>(out, W2, h, n);
    k_self_init   <<<gT,    B, 0, stream>>>(dis, h, agg, total);
    k_edge_scatter<<<gSc,   B, 0, stream>>>(ei, ew, dis, h, agg, E);
    k_final       <<<gT,    B, 0, stream>>>(agg, b2, out, total);
}